// InterferenceGNN_22806276342401
// MI455X (gfx1250) — compile-verified
//
#include <hip/hip_runtime.h>

#define N_NODES 100000
#define N_EDGES 2000000
#define IN_CH 6
#define HID 32
#define BN_EPS 1e-5f

typedef float v2f __attribute__((ext_vector_type(2)));
typedef float v8f __attribute__((ext_vector_type(8)));

// ---------------------------------------------------------------------------
// agg initialization: -inf bit pattern (identity for max)
// ---------------------------------------------------------------------------
__global__ void fill_neg_inf_kernel(unsigned int* __restrict__ p, int n) {
  int i = blockIdx.x * blockDim.x + threadIdx.x;
  if (i < n) p[i] = 0xFF800000u;  // -inf
}

// float atomic max via signed-int max (v>=0) / unsigned-int min (v<0).
// Correct for all finite-value orderings with -inf init; idempotent.
__device__ __forceinline__ void atomic_max_f32(float* addr, float v) {
  if (v >= 0.0f) atomicMax((int*)addr, __float_as_int(v));
  else           atomicMin((unsigned int*)addr, __float_as_uint(v));
}

// ---------------------------------------------------------------------------
// Edge kernel: one wave32 = one 16-edge M-tile.
//   GEMM1: [16 x 2d] @ [2d x 32]  via v_wmma_f32_16x16x4_f32 (K/4 steps, 2 N-tiles)
//   BN(eval)+ReLU folded to per-column scale/shift
//   GEMM2: [16 x 32] @ [32 x 32]  (8 K-steps, 2 N-tiles)
//   scatter-max into agg[dst]
// ---------------------------------------------------------------------------
#define S1 68   // LDS row stride for stage-1 A matrix (16 x 2d), conflict-free
#define S2 34   // LDS row stride for stage-2 A matrix (16 x 32), conflict-free

__launch_bounds__(256)
__global__ void edgeconv_kernel(const float* __restrict__ h, int dch,
                                const int* __restrict__ src,
                                const int* __restrict__ dst,
                                const float* __restrict__ W1,
                                const float* __restrict__ b1,
                                const float* __restrict__ gamma,
                                const float* __restrict__ beta,
                                const float* __restrict__ mu,
                                const float* __restrict__ var,
                                const float* __restrict__ W2,
                                const float* __restrict__ b2,
                                float* __restrict__ agg, int E) {
  __shared__ float smem[8 * 16 * S1];          // 34816 B: per-wave tile staging
  const int lane  = threadIdx.x & 31;
  const int wave  = threadIdx.x >> 5;
  const int nhalf = lane >> 4;                 // 0: lanes 0-15, 1: lanes 16-31
  const int nl    = lane & 15;
  float* lds = smem + wave * (16 * S1);

  const int e0 = (blockIdx.x * 8 + wave) * 16;

  // ---- stage A = [hi | hj - hi]  (rows 0..15, K = 2*dch), lanes 0-15 only ----
  int de = 0;
  if (lane < 16) {
    int e = e0 + lane;
    if (e >= E) e = E - 1;                     // clamp: max-scatter is idempotent
    de = dst[e];
    const int se = src[e];
    const float* hi = h + (size_t)de * dch;
    const float* hj = h + (size_t)se * dch;
    float* row = lds + lane * S1;
    for (int k = 0; k < dch; ++k) {
      float a = hi[k];
      row[k]       = a;
      row[dch + k] = hj[k] - a;
    }
  }
  // Same-wave LDS producer/consumer: DS ops are in-order; compiler inserts
  // s_wait_dscnt for the RAW. No barrier needed (EXEC must be all-1 for WMMA).

  // ---- GEMM1: K1 = 2*dch, N-tiles {0,1} ----
  const int K1 = 2 * dch;
  v8f c0 = {};
  v8f c1 = {};
  for (int kk = 0; kk < K1; kk += 4) {
    const int kb = kk + nhalf * 2;             // A/B K-pair per lane half
    v2f a;  a.x = lds[nl * S1 + kb];     a.y = lds[nl * S1 + kb + 1];
    v2f bA; bA.x = W1[kb * HID + nl];          bA.y = W1[(kb + 1) * HID + nl];
    v2f bB; bB.x = W1[kb * HID + nl + 16];     bB.y = W1[(kb + 1) * HID + nl + 16];
    c0 = __builtin_amdgcn_wmma_f32_16x16x4_f32(false, a, false, bA, (short)0, c0, false, false);
    c1 = __builtin_amdgcn_wmma_f32_16x16x4_f32(false, a, false, bB, (short)0, c1, false, false);
  }

  // ---- BN(eval) + bias + ReLU, fused per column: m = relu(m0*sc + sh) ----
  float sc0, sh0, sc1, sh1;
  {
    int n = nl;
    float s = gamma[n] * rsqrtf(var[n] + BN_EPS);
    sc0 = s; sh0 = (b1[n] - mu[n]) * s + beta[n];
    n = nl + 16;
    s = gamma[n] * rsqrtf(var[n] + BN_EPS);
    sc1 = s; sh1 = (b1[n] - mu[n]) * s + beta[n];
  }
  // restage as A-matrix for GEMM2 (row = M, col = K), stride S2
  for (int i = 0; i < 8; ++i) {
    const int rowM = i + nhalf * 8;            // C layout: VGPR i -> M = i + 8*half
    lds[rowM * S2 + nl]      = fmaxf(fmaf(c0[i], sc0, sh0), 0.0f);
    lds[rowM * S2 + nl + 16] = fmaxf(fmaf(c1[i], sc1, sh1), 0.0f);
  }

  // ---- GEMM2: K = 32, N-tiles {0,1} ----
  v8f d0 = {};
  v8f d1 = {};
  for (int kk = 0; kk < HID; kk += 4) {
    const int kb = kk + nhalf * 2;
    v2f a;  a.x = lds[nl * S2 + kb];     a.y = lds[nl * S2 + kb + 1];
    v2f bA; bA.x = W2[kb * HID + nl];          bA.y = W2[(kb + 1) * HID + nl];
    v2f bB; bB.x = W2[kb * HID + nl + 16];     bB.y = W2[(kb + 1) * HID + nl + 16];
    d0 = __builtin_amdgcn_wmma_f32_16x16x4_f32(false, a, false, bA, (short)0, d0, false, false);
    d1 = __builtin_amdgcn_wmma_f32_16x16x4_f32(false, a, false, bB, (short)0, d1, false, false);
  }

  // ---- + b2, scatter-max into agg[dst] ----
  const float bb0 = b2[nl];
  const float bb1 = b2[nl + 16];
  for (int i = 0; i < 8; ++i) {
    const int srcLane = (lane < 16) ? i : (i + 8);   // row M of this C element
    const int idx = __shfl(de, srcLane, 32);
    float* row = agg + (size_t)idx * HID;
    atomic_max_f32(row + nl,      d0[i] + bb0);
    atomic_max_f32(row + nl + 16, d1[i] + bb1);
  }
}

// ---------------------------------------------------------------------------
// Node kernel: isolated (-inf) -> 0, then ELU. Safe in-place.
// ---------------------------------------------------------------------------
__global__ void node_finish_kernel(const float* __restrict__ agg,
                                   float* __restrict__ out, int n) {
  int i = blockIdx.x * blockDim.x + threadIdx.x;
  if (i >= n) return;
  float v = agg[i];
  if (!(fabsf(v) < 3.0e38f)) v = 0.0f;         // -inf (empty segment) -> 0
  out[i] = (v > 0.0f) ? v : expm1f(v);         // ELU
}

// ---------------------------------------------------------------------------
// Launch: 3 layers, each = fill(-inf) -> edge conv/scatter -> finish(ELU).
// ws layout: bufA [N*32], bufB [N*32]  (25.6 MB)
// ---------------------------------------------------------------------------
extern "C" void kernel_launch(void* const* d_in, const int* in_sizes, int n_in,
                              void* d_out, int out_size, void* d_ws, size_t ws_size,
                              hipStream_t stream) {
  (void)in_sizes; (void)n_in; (void)out_size; (void)ws_size;

  const float* x   = (const float*)d_in[0];
  const int*   ei  = (const int*)d_in[1];
  const int*   src = ei;             // edge_index[0]  (x_j, neighbor)
  const int*   dst = ei + N_EDGES;   // edge_index[1]  (x_i, aggregation target)

  float* bufA = (float*)d_ws;
  float* bufB = bufA + (size_t)N_NODES * HID;

  const int NC = N_NODES * HID;
  const dim3 blk(256);
  const dim3 grdN((NC + 255) / 256);
  const dim3 grdE((N_EDGES + 127) / 128);      // 8 waves x 16 edges per block

  const float* hin = x;
  int dch = IN_CH;
  for (int l = 0; l < 3; ++l) {
    const float* W1 = (const float*)d_in[2 + l * 8 + 0];
    const float* b1 = (const float*)d_in[2 + l * 8 + 1];
    const float* ga = (const float*)d_in[2 + l * 8 + 2];
    const float* be = (const float*)d_in[2 + l * 8 + 3];
    const float* mu = (const float*)d_in[2 + l * 8 + 4];
    const float* va = (const float*)d_in[2 + l * 8 + 5];
    const float* W2 = (const float*)d_in[2 + l * 8 + 6];
    const float* b2 = (const float*)d_in[2 + l * 8 + 7];

    float* agg  = (l == 1) ? bufB : bufA;      // ping-pong; hin never aliases agg
    float* hout = (l == 2) ? (float*)d_out : agg;

    fill_neg_inf_kernel<<<grdN, blk, 0, stream>>>((unsigned int*)agg, NC);
    edgeconv_kernel<<<grdE, blk, 0, stream>>>(hin, dch, src, dst,
                                              W1, b1, ga, be, mu, va, W2, b2,
                                              agg, N_EDGES);
    node_finish_kernel<<<grdN, blk, 0, stream>>>(agg, hout, NC);

    hin = hout;
    dch = HID;
  }
}